// Autoencoder_22789096472707
// MI455X (gfx1250) — compile-verified
//
#include <hip/hip_runtime.h>
#include <math.h>

typedef __attribute__((ext_vector_type(2))) float v2f;
typedef __attribute__((ext_vector_type(8))) float v8f;

// ---------------- degree / normalization ----------------

__global__ void k_deg_init(float* __restrict__ deg, int n) {
  int i = blockIdx.x * blockDim.x + threadIdx.x;
  if (i < n) deg[i] = 1.0f;  // self-loop contributes 1 to every node's degree
}

__global__ void k_deg_edges(const int* __restrict__ dst, float* __restrict__ deg, int nE) {
  int e = blockIdx.x * blockDim.x + threadIdx.x;
  if (e < nE) atomicAdd(&deg[dst[e]], 1.0f);
}

__global__ void k_deg_rsqrt(float* __restrict__ deg, int n) {
  int i = blockIdx.x * blockDim.x + threadIdx.x;
  if (i < n) deg[i] = rsqrtf(deg[i]);  // deg >= 1 always (self-loops)
}

// ---------------- per-layer linear via V_WMMA_F32_16X16X4_F32 ----------------
// y[i,:] = (h[i,:] @ W) * dis[i]   (dis[src] folded in)
// s      = copy of y               (self-loop seed for the scatter accumulator)
// One wave32 computes a 16-node x FOUT tile: A = 16x4 fp32 chunk of h,
// B = 4x16 fp32 chunk of W (columns >= FOUT zero-padded), C/D = 16x16 fp32.
template <int FIN, int FOUT>
__global__ __launch_bounds__(128) void k_linear_wmma(
    const float* __restrict__ h, const float* __restrict__ W,
    const float* __restrict__ dis, float* __restrict__ y, float* __restrict__ s,
    int nNodes) {
  const int lane = threadIdx.x & 31;
  const int wave = threadIdx.x >> 5;
  const int tile = blockIdx.x * 4 + wave;
  const int base = tile * 16;
  if (base >= nNodes) return;  // uniform per wave -> EXEC stays all-1s

  const int half = lane >> 4;   // 0: K pair {0,1}; 1: K pair {2,3}
  const int lid  = lane & 15;   // A: row M; B: column N
  const int m = base + lid;

  v8f acc = {};
#pragma unroll
  for (int k0 = 0; k0 < FIN; k0 += 4) {
    const int ka = k0 + half * 2;
    v2f a, b;
    // A 16x4 fp32 layout: lanes 0-15 hold (K=k0,k0+1), lanes 16-31 (K=k0+2,k0+3)
    a.x = h[m * FIN + ka + 0];
    a.y = h[m * FIN + ka + 1];
    // B 4x16 fp32: lane gives column N, halves give K pairs (mirror of A)
    b.x = (lid < FOUT) ? W[(ka + 0) * FOUT + lid] : 0.0f;
    b.y = (lid < FOUT) ? W[(ka + 1) * FOUT + lid] : 0.0f;
    acc = __builtin_amdgcn_wmma_f32_16x16x4_f32(
        /*neg_a=*/false, a, /*neg_b=*/false, b,
        /*c_mod=*/(short)0, acc, /*reuse_a=*/false, /*reuse_b=*/false);
  }

  // C/D layout: VGPR r holds M=r (lanes 0-15) / M=r+8 (lanes 16-31), N=lid
  if (lid < FOUT) {
#pragma unroll
    for (int r = 0; r < 8; ++r) {
      const int row = base + r + half * 8;
      const float v = acc[r] * dis[row];
      y[row * FOUT + lid] = v;  // gather source for the scatter pass
      s[row * FOUT + lid] = v;  // accumulator seeded with the self-loop term
    }
  }
}

// ---------------- edge scatter: s[dst] += y[src] ----------------
// Coalesced int32 index stream from HBM; y gathers and s atomics hit the
// 192 MB L2 (both buffers are ~12.8 MB).
template <int F>
__global__ void k_scatter(const float* __restrict__ y, float* __restrict__ s,
                          const int* __restrict__ src, const int* __restrict__ dst,
                          int nE) {
  int e = blockIdx.x * blockDim.x + threadIdx.x;
  if (e >= nE) return;
  const int si = src[e];
  const int di = dst[e];
  const float4* yv = (const float4*)(y + (size_t)si * F);
  float* sp = s + (size_t)di * F;
#pragma unroll
  for (int q = 0; q < F / 4; ++q) {
    float4 v = yv[q];
    atomicAdd(sp + 4 * q + 0, v.x);
    atomicAdd(sp + 4 * q + 1, v.y);
    atomicAdd(sp + 4 * q + 2, v.z);
    atomicAdd(sp + 4 * q + 3, v.w);
  }
}

// ---------------- finalize: out = act(dis[i]*s[i,:] + b) ----------------
// ACT: 0 = identity, 1 = relu, 2 = sigmoid
template <int F, int ACT>
__global__ void k_finalize(const float* __restrict__ s, const float* __restrict__ dis,
                           const float* __restrict__ bias, float* __restrict__ out,
                           int n) {
  int t = blockIdx.x * blockDim.x + threadIdx.x;
  if (t >= n * F) return;
  const int i = t / F;
  const int c = t - i * F;
  float v = s[t] * dis[i] + bias[c];
  if (ACT == 1) v = fmaxf(v, 0.0f);
  else if (ACT == 2) v = 1.0f / (1.0f + __expf(-v));
  out[t] = v;
}

// ---------------- host orchestration ----------------

static inline size_t align256(size_t x) { return (x + 255) & ~(size_t)255; }

extern "C" void kernel_launch(void* const* d_in, const int* in_sizes, int n_in,
                              void* d_out, int out_size, void* d_ws, size_t ws_size,
                              hipStream_t stream) {
  (void)n_in; (void)out_size; (void)ws_size;

  const float* x  = (const float*)d_in[0];
  const int*   ei = (const int*)d_in[1];   // edge_index [2,E] flat, int32
  const float* W1 = (const float*)d_in[2];
  const float* b1 = (const float*)d_in[3];
  const float* W2 = (const float*)d_in[4];
  const float* b2 = (const float*)d_in[5];
  const float* W3 = (const float*)d_in[6];
  const float* b3 = (const float*)d_in[7];
  const float* W4 = (const float*)d_in[8];
  const float* b4 = (const float*)d_in[9];
  float* out = (float*)d_out;

  const int N = in_sizes[0] / 12;
  const int E = in_sizes[1] / 2;
  const int* src = ei;
  const int* dst = ei + E;

  // workspace: dis[N] + three N x 16 ping-pong buffers (~39.2 MB total)
  char* ws = (char*)d_ws;
  size_t off = 0;
  float* dis  = (float*)(ws + off); off += align256((size_t)N * 4);
  float* buf0 = (float*)(ws + off); off += align256((size_t)N * 16 * 4);
  float* buf1 = (float*)(ws + off); off += align256((size_t)N * 16 * 4);
  float* buf2 = (float*)(ws + off); off += align256((size_t)N * 16 * 4);

  const int TB = 256;
  const dim3 gN((N + TB - 1) / TB), gE((E + TB - 1) / TB), tb(TB);
  const int tiles = (N + 15) / 16;
  const dim3 gL((tiles + 3) / 4), tbL(128);

  // symmetric normalization: dis = rsqrt(1 + in-degree)
  k_deg_init<<<gN, tb, 0, stream>>>(dis, N);
  k_deg_edges<<<gE, tb, 0, stream>>>(dst, dis, E);
  k_deg_rsqrt<<<gN, tb, 0, stream>>>(dis, N);

  // Layer 1: 12 -> 16, relu
  k_linear_wmma<12, 16><<<gL, tbL, 0, stream>>>(x, W1, dis, buf0, buf1, N);
  k_scatter<16><<<gE, tb, 0, stream>>>(buf0, buf1, src, dst, E);
  k_finalize<16, 1><<<(N * 16 + TB - 1) / TB, tb, 0, stream>>>(buf1, dis, b1, buf2, N);

  // Layer 2: 16 -> 8, identity
  k_linear_wmma<16, 8><<<gL, tbL, 0, stream>>>(buf2, W2, dis, buf0, buf1, N);
  k_scatter<8><<<gE, tb, 0, stream>>>(buf0, buf1, src, dst, E);
  k_finalize<8, 0><<<(N * 8 + TB - 1) / TB, tb, 0, stream>>>(buf1, dis, b2, buf2, N);

  // Layer 3: 8 -> 16, relu
  k_linear_wmma<8, 16><<<gL, tbL, 0, stream>>>(buf2, W3, dis, buf0, buf1, N);
  k_scatter<16><<<gE, tb, 0, stream>>>(buf0, buf1, src, dst, E);
  k_finalize<16, 1><<<(N * 16 + TB - 1) / TB, tb, 0, stream>>>(buf1, dis, b3, buf2, N);

  // Layer 4: 16 -> 12, sigmoid -> d_out
  k_linear_wmma<16, 12><<<gL, tbL, 0, stream>>>(buf2, W4, dis, buf0, buf1, N);
  k_scatter<12><<<gE, tb, 0, stream>>>(buf0, buf1, src, dst, E);
  k_finalize<12, 2><<<(N * 12 + TB - 1) / TB, tb, 0, stream>>>(buf1, dis, b4, out, N);
}